// SparseBoundaryContent_8237747274016
// MI455X (gfx1250) — compile-verified
//
#include <hip/hip_runtime.h>
#include <cstdint>

#define NCLIP 64
typedef __attribute__((ext_vector_type(4))) float v4f;

// Closed-form mask for POOLING_COUNTS = [15, 8, 8], N = 64.
//  d = j - i:  0..15 -> stride-1 lines (+diag)
//              17,19,...,31 (odd) with even i -> stride-2 lines
//              35,39,...,63 (d%4==3) with i%4==0 -> stride-4 lines
__device__ __forceinline__ bool is_masked(int i, int j) {
    int d = j - i;
    if (d < 0) return false;
    if (d <= 15) return true;
    if (d <= 31) return (d >= 17) && (d & 1) && !(i & 1);
    if (d >= 35) return ((d & 3) == 3) && !(i & 3);
    return false;
}

__global__ __launch_bounds__(256) void bcmap_kernel(const float* __restrict__ x,
                                                    float* __restrict__ out,
                                                    int BD) {
    __shared__ float T[7][NCLIP];   // sparse range-max table, T[0] = x row
    const int tid = threadIdx.x;
    const int r   = blockIdx.x;     // r = b*D + d

    // ---- gfx1250 async global -> LDS copy of the 64-float row (wave 0) ----
    // 32 lanes x 8 bytes = 256 bytes. LDS dest address = low 32 bits of the
    // flat shared pointer (aperture-relative LDS byte offset).
    if (tid < 32) {
        unsigned lds_off = (unsigned)(uintptr_t)(&T[0][0]) + (unsigned)tid * 8u;
        uint64_t gaddr   = (uint64_t)(uintptr_t)(x + (size_t)r * NCLIP) + (uint64_t)tid * 8u;
        asm volatile("global_load_async_to_lds_b64 %0, %1, off"
                     :
                     : "v"(lds_off), "v"(gaddr)
                     : "memory");
        asm volatile("s_wait_asynccnt 0" ::: "memory");
    }
    __syncthreads();

    // ---- build sparse table: T[k][i] = max(x[i .. i + 2^k - 1]) ----
    #pragma unroll
    for (int k = 1; k < 7; ++k) {
        if (tid < NCLIP) {
            int step = 1 << (k - 1);
            int i2 = tid + step;
            if (i2 > NCLIP - 1) i2 = NCLIP - 1;   // clamp; clamped entries never queried
            T[k][tid] = fmaxf(T[k - 1][tid], T[k - 1][i2]);
        }
        __syncthreads();
    }

    // ---- emit boundary + content tiles (4096 floats each, float4 NT stores) ----
    float* bout = out + (size_t)r * (NCLIP * NCLIP);
    float* cout = bout + (size_t)BD * (NCLIP * NCLIP);

    for (int g = tid; g < (NCLIP * NCLIP / 4); g += 256) {
        const int i  = g >> 4;           // row of the 64x64 tile
        const int j0 = (g & 15) << 2;    // first of 4 consecutive cols
        const float xi = T[0][i];
        v4f bv, cv;
        #pragma unroll
        for (int l = 0; l < 4; ++l) {
            const int j = j0 + l;
            float b = 0.0f, c = 0.0f;
            if (is_masked(i, j)) {
                b = 0.5f * (xi + T[0][j]);
                const int len = j - i + 1;
                const int k = 31 - __clz(len);           // floor(log2(len))
                c = fmaxf(T[k][i], T[k][j - (1 << k) + 1]);
            }
            bv[l] = b;
            cv[l] = c;
        }
        __builtin_nontemporal_store(bv, (v4f*)(bout + (size_t)g * 4));
        __builtin_nontemporal_store(cv, (v4f*)(cout + (size_t)g * 4));
    }
}

__global__ __launch_bounds__(256) void bcmask_kernel(float* __restrict__ m, int total) {
    int idx = blockIdx.x * 256 + threadIdx.x;
    if (idx < total) {
        int w = idx & (NCLIP * NCLIP - 1);    // mask tile replicated over B
        int i = w >> 6;
        int j = w & (NCLIP - 1);
        m[idx] = is_masked(i, j) ? 1.0f : 0.0f;
    }
}

extern "C" void kernel_launch(void* const* d_in, const int* in_sizes, int n_in,
                              void* d_out, int out_size, void* d_ws, size_t ws_size,
                              hipStream_t stream) {
    (void)n_in; (void)d_ws; (void)ws_size;
    const float* x = (const float*)d_in[0];
    float* out = (float*)d_out;

    const int BD = in_sizes[0] / NCLIP;                 // B*D = 8192
    // boundary | content | mask, concatenated flat in return order
    bcmap_kernel<<<BD, 256, 0, stream>>>(x, out, BD);

    const long long mask_off = 2LL * BD * NCLIP * NCLIP;
    const int mask_total = out_size - (int)mask_off;    // B * 64 * 64
    if (mask_total > 0) {
        int blocks = (mask_total + 255) / 256;
        bcmask_kernel<<<blocks, 256, 0, stream>>>(out + mask_off, mask_total);
    }
}